// GumbelTopK_75943611727994
// MI455X (gfx1250) — compile-verified
//
#include <hip/hip_runtime.h>
#include <hip/hip_bf16.h>
#include <stdint.h>

// ---------------------------------------------------------------------------
// Gumbel top-k mask, MI455X (gfx1250).
// Memory-bound (~575 MB total traffic -> ~25us @ 23.3 TB/s). No matrix math,
// so the CDNA5 paths used are: TDM tensor_load_to_lds async streaming with
// s_wait_tensorcnt, LDS ds_add histograms, and non-temporal B128 global I/O.
// ---------------------------------------------------------------------------

typedef float    f32x4 __attribute__((ext_vector_type(4)));
typedef uint32_t u32x4 __attribute__((ext_vector_type(4)));
typedef int      i32x4 __attribute__((ext_vector_type(4)));
typedef int      i32x8 __attribute__((ext_vector_type(8)));

#define GTK_EPS 1e-8f
#define RTILE   512u   // words per TDM chunk (2 KB)

__device__ __forceinline__ float gtk_gumbel(float u) {
    return -__logf(-__logf(u + GTK_EPS) + GTK_EPS);
}

// monotonic float -> uint key (larger float => larger key)
__device__ __forceinline__ uint32_t gtk_f2key(float f) {
    uint32_t b = __float_as_uint(f);
    return (b & 0x80000000u) ? ~b : (b | 0x80000000u);
}

// ---------------------------------------------------------------------------
// Pass 0: zero histograms + state (ws is poisoned 0xAA; must re-init per call)
// ---------------------------------------------------------------------------
__global__ void gtk_init(uint32_t* hist0, uint32_t* hist1, uint32_t* hist2,
                         uint32_t* st) {
    uint32_t i = blockIdx.x * blockDim.x + threadIdx.x;
    if (i < 4096) hist0[i] = 0;
    if (i < 4096) hist1[i] = 0;
    if (i < 256)  hist2[i]  = 0;
    if (i < 8)    st[i]     = 0;  // st[6] = tie counter
}

// ---------------------------------------------------------------------------
// Pass 1: keys[i] = sortable(mask_logits + gumbel(u1)); histogram top 12 bits.
// Inputs are stream-once -> non-temporal loads (don't pollute L2; keys must
// stay L2-resident: 64 MB < 192 MB).
// ---------------------------------------------------------------------------
__global__ void gtk_build(const float* __restrict__ logits,
                          const float* __restrict__ u1,
                          const int* __restrict__ training,
                          uint32_t* __restrict__ keys,
                          uint32_t* __restrict__ hist0,
                          uint32_t n4, uint32_t N) {
    __shared__ uint32_t lh[4096];
    for (uint32_t i = threadIdx.x; i < 4096; i += blockDim.x) lh[i] = 0;
    __syncthreads();

    const bool tr = (*training) != 0;
    const uint32_t stride = gridDim.x * blockDim.x;
    for (uint32_t i = blockIdx.x * blockDim.x + threadIdx.x; i < n4; i += stride) {
        f32x4 L = __builtin_nontemporal_load(((const f32x4*)logits) + i);
        f32x4 U = __builtin_nontemporal_load(((const f32x4*)u1) + i);
        u32x4 kv;
#pragma unroll
        for (int c = 0; c < 4; ++c) {
            float v = tr ? (L[c] + gtk_gumbel(U[c])) : L[c];
            kv[c] = gtk_f2key(v);
            atomicAdd(&lh[kv[c] >> 20], 1u);  // ds_add_u32
        }
        *(((u32x4*)keys) + i) = kv;           // regular store: keep in L2
    }
    // scalar tail (N not multiple of 4)
    if (blockIdx.x == 0 && threadIdx.x == 0) {
        for (uint32_t i = n4 * 4u; i < N; ++i) {
            float v = tr ? (logits[i] + gtk_gumbel(u1[i])) : logits[i];
            uint32_t kk = gtk_f2key(v);
            keys[i] = kk;
            atomicAdd(&hist0[kk >> 20], 1u);
        }
    }
    __syncthreads();
    for (uint32_t i = threadIdx.x; i < 4096; i += blockDim.x) {
        uint32_t c = lh[i];
        if (c) atomicAdd(&hist0[i], c);
    }
}

// ---------------------------------------------------------------------------
// Scan (1 thread): walk histogram top-down, find bucket crossing k-remaining.
// st[2*level] = selected bucket, st[2*level+1] = remaining count inside it.
// ---------------------------------------------------------------------------
__global__ void gtk_scan(const uint32_t* __restrict__ hist, int bins,
                         const int* __restrict__ kptr, uint32_t* st, int level) {
    uint32_t kr = (level == 0) ? (uint32_t)(*kptr) : st[2 * level - 1];
    uint32_t cum = 0, sel = 0, rem = kr;
    for (int b = bins - 1; b >= 0; --b) {
        uint32_t c = hist[b];
        if (cum + c >= kr) { sel = (uint32_t)b; rem = kr - cum; break; }
        cum += c;
    }
    st[2 * level]     = sel;
    st[2 * level + 1] = rem;
}

// ---------------------------------------------------------------------------
// TDM: async DMA one chunk of keys (RTILE dwords) global -> LDS.
// D# group0: count=1, lds_addr, global_addr[56:0], type=2.
// D# group1: data_size=4B; 2D tile 512x1; tensor dims match tile (no OOB).
// 6-arg clang-23 builtin form (probe: this toolchain expects 6 args).
// ---------------------------------------------------------------------------
__device__ __forceinline__ void gtk_tdm_load(const uint32_t* gsrc,
                                             uint32_t lds_off, uint32_t words) {
    uint64_t ga = (uint64_t)(uintptr_t)gsrc;
    u32x4 g0;
    g0[0] = 1u;                                              // count=1, user
    g0[1] = lds_off;                                         // lds_addr
    g0[2] = (uint32_t)ga;                                    // global_addr lo
    g0[3] = (uint32_t)((ga >> 32) & 0x01FFFFFFu) | (2u << 30); // hi + type=2
    i32x8 g1;
    g1[0] = (int)(2u << 16);                                 // data_size=4B
    g1[1] = (int)((words & 0xFFFFu) << 16);                  // tensor_dim0 lo16
    g1[2] = (int)(((words >> 16) & 0xFFFFu) | (1u << 16));   // dim0 hi, dim1=1
    g1[3] = (int)((words & 0xFFFFu) << 16);                  // tile_dim0
    g1[4] = 1;                                               // tile_dim1=1
    g1[5] = (int)words;                                      // dim0_stride
    g1[6] = 0;
    g1[7] = 0;
    i32x4 z4 = (i32x4){0, 0, 0, 0};
    i32x8 z8 = (i32x8){0, 0, 0, 0, 0, 0, 0, 0};
    __builtin_amdgcn_tensor_load_to_lds(g0, g1, z4, z4, z8, 0);
}

// ---------------------------------------------------------------------------
// Passes 2 & 3: refine histogram among keys matching the already-selected
// high bits. Each wave streams chunks via TDM into its private double buffer
// (per-wave TENSORcnt, no cross-wave sync needed), reads LDS via B128.
// ---------------------------------------------------------------------------
__global__ void gtk_refine(const uint32_t* __restrict__ keys, uint32_t N,
                           uint32_t* __restrict__ ghist,
                           const uint32_t* __restrict__ st, int level) {
    __shared__ uint32_t lhist[4096];
    __shared__ uint32_t sbuf[8 * 2 * RTILE];   // 8 waves x 2 buffers x 2 KB

    uint32_t bins, match_shift, match_val, sub_shift, sub_mask;
    if (level == 1) {
        bins = 4096; match_shift = 20; match_val = st[0];
        sub_shift = 8; sub_mask = 0xFFFu;
    } else {
        bins = 256;  match_shift = 8;  match_val = (st[0] << 12) | st[2];
        sub_shift = 0; sub_mask = 0xFFu;
    }
    for (uint32_t i = threadIdx.x; i < bins; i += blockDim.x) lhist[i] = 0;
    __syncthreads();

    const uint32_t lane   = threadIdx.x & 31u;
    const uint32_t wave   = threadIdx.x >> 5;
    const uint32_t wpb    = blockDim.x >> 5;
    const uint32_t gwave  = blockIdx.x * wpb + wave;
    const uint32_t nwaves = gridDim.x * wpb;
    const uint32_t nchunks = N / RTILE;

    uint32_t* buf0 = &sbuf[wave * 2u * RTILE];
    uint32_t* buf1 = buf0 + RTILE;
    uint32_t  off0 = (uint32_t)(uintptr_t)buf0;   // LDS byte offset
    uint32_t  off1 = (uint32_t)(uintptr_t)buf1;

    uint32_t it = 0;
    for (uint32_t c = gwave; c < nchunks; c += nwaves, ++it) {
        uint32_t cur = it & 1u;
        if (it == 0)
            gtk_tdm_load(keys + (uint64_t)c * RTILE, cur ? off1 : off0, RTILE);
        uint32_t cn = c + nwaves;
        bool more = (cn < nchunks);
        if (more) {
            // previous reads of the other buffer must land before TDM rewrites
            asm volatile("s_wait_dscnt 0x0" ::: "memory");
            gtk_tdm_load(keys + (uint64_t)cn * RTILE, cur ? off0 : off1, RTILE);
        }
        if (more) __builtin_amdgcn_s_wait_tensorcnt(1);
        else      __builtin_amdgcn_s_wait_tensorcnt(0);

        const u32x4* p = (const u32x4*)(cur ? buf1 : buf0);
#pragma unroll
        for (uint32_t j = 0; j < RTILE / 4u / 32u; ++j) {   // ds_load_b128
            u32x4 w = p[lane + 32u * j];
#pragma unroll
            for (int x = 0; x < 4; ++x) {
                uint32_t kk = w[x];
                if ((kk >> match_shift) == match_val)
                    atomicAdd(&lhist[(kk >> sub_shift) & sub_mask], 1u);
            }
        }
    }
    // tail (N not multiple of RTILE)
    if (gwave == 0) {
        for (uint32_t i = nchunks * RTILE + lane; i < N; i += 32u) {
            uint32_t kk = keys[i];
            if ((kk >> match_shift) == match_val)
                atomicAdd(&lhist[(kk >> sub_shift) & sub_mask], 1u);
        }
    }
    __syncthreads();
    for (uint32_t i = threadIdx.x; i < bins; i += blockDim.x) {
        uint32_t cv = lhist[i];
        if (cv) atomicAdd(&ghist[i], cv);
    }
}

// ---------------------------------------------------------------------------
// Pass 4: select + straight-through gumbel-softmax forward value.
// out = training ? ((mask + g(u2[:,1]) > g(u2[:,0])) ? 1 : 0) : mask
// ---------------------------------------------------------------------------
__global__ void gtk_final(const uint32_t* __restrict__ keys,
                          const float* __restrict__ u2,
                          const int* __restrict__ training,
                          const uint32_t* __restrict__ st,
                          uint32_t* __restrict__ tie,
                          float* __restrict__ out, uint32_t n4, uint32_t N) {
    const uint32_t T   = (st[0] << 20) | (st[2] << 8) | st[4];
    const uint32_t rem = st[5];
    const bool tr = (*training) != 0;
    const uint32_t stride = gridDim.x * blockDim.x;
    for (uint32_t i = blockIdx.x * blockDim.x + threadIdx.x; i < n4; i += stride) {
        u32x4 kv = __builtin_nontemporal_load(((const u32x4*)keys) + i);  // last use
        f32x4 a  = __builtin_nontemporal_load(((const f32x4*)u2) + 2 * i);
        f32x4 b  = __builtin_nontemporal_load(((const f32x4*)u2) + 2 * i + 1);
        f32x4 o;
#pragma unroll
        for (int c = 0; c < 4; ++c) {
            uint32_t kk = kv[c];
            float m;
            if (kk > T)       m = 1.0f;
            else if (kk == T) m = (atomicAdd(tie, 1u) < rem) ? 1.0f : 0.0f;
            else              m = 0.0f;
            float g0u = (c < 2) ? a[2 * c]     : b[2 * (c - 2)];
            float g1u = (c < 2) ? a[2 * c + 1] : b[2 * (c - 2) + 1];
            if (tr) {
                float g0 = gtk_gumbel(g0u), g1 = gtk_gumbel(g1u);
                o[c] = (m + g1 > g0) ? 1.0f : 0.0f;
            } else {
                o[c] = m;
            }
        }
        __builtin_nontemporal_store(o, ((f32x4*)out) + i);
    }
    if (blockIdx.x == 0 && threadIdx.x == 0) {
        for (uint32_t i = n4 * 4u; i < N; ++i) {
            uint32_t kk = keys[i];
            float m = (kk > T) ? 1.0f
                    : (kk == T ? ((atomicAdd(tie, 1u) < rem) ? 1.0f : 0.0f) : 0.0f);
            if (tr) {
                float g0 = gtk_gumbel(u2[2 * i]), g1 = gtk_gumbel(u2[2 * i + 1]);
                out[i] = (m + g1 > g0) ? 1.0f : 0.0f;
            } else out[i] = m;
        }
    }
}

// ---------------------------------------------------------------------------
extern "C" void kernel_launch(void* const* d_in, const int* in_sizes, int n_in,
                              void* d_out, int out_size, void* d_ws, size_t ws_size,
                              hipStream_t stream) {
    (void)n_in; (void)out_size; (void)ws_size;
    const float* logits = (const float*)d_in[0];
    const float* u1     = (const float*)d_in[1];
    const float* u2     = (const float*)d_in[2];
    const int*   kptr   = (const int*)d_in[3];
    const int*   trptr  = (const int*)d_in[4];
    float*       out    = (float*)d_out;

    const uint32_t N  = (uint32_t)in_sizes[0];
    const uint32_t n4 = N / 4u;

    // workspace: keys[N] | hist0[4096] | hist1[4096] | hist2[256] | st[8]
    uint32_t* keys  = (uint32_t*)d_ws;
    uint32_t* hist0 = keys + N;
    uint32_t* hist1 = hist0 + 4096;
    uint32_t* hist2 = hist1 + 4096;
    uint32_t* st    = hist2 + 256;
    uint32_t* tie   = st + 6;

    gtk_init  <<<16,   256, 0, stream>>>(hist0, hist1, hist2, st);
    gtk_build <<<2048, 256, 0, stream>>>(logits, u1, trptr, keys, hist0, n4, N);
    gtk_scan  <<<1,    1,   0, stream>>>(hist0, 4096, kptr, st, 0);
    gtk_refine<<<2048, 256, 0, stream>>>(keys, N, hist1, st, 1);
    gtk_scan  <<<1,    1,   0, stream>>>(hist1, 4096, kptr, st, 1);
    gtk_refine<<<2048, 256, 0, stream>>>(keys, N, hist2, st, 2);
    gtk_scan  <<<1,    1,   0, stream>>>(hist2, 256, kptr, st, 2);
    gtk_final <<<2048, 256, 0, stream>>>(keys, u2, trptr, st, tie, out, n4, N);
}